// SoftPatternClassifier_85203561218539
// MI455X (gfx1250) — compile-verified
//
#include <hip/hip_runtime.h>
#include <math.h>

// ---------------------------------------------------------------------------
// SoftPatternClassifier fused kernel for MI455X (gfx1250, wave32, WMMA).
//   Stage 1 (fused): per (doc, 16-pattern tile) block:
//       logits tile via V_WMMA_F32_16X16X4_F32  ->  sigmoid(.+bias) in LDS
//       ->  max-plus scan (h state in registers across all 128 tokens)
//       ->  scores (64,256) in d_ws
//   Stage 2: tiny MLP head + log_softmax -> d_out (64,2)
// ---------------------------------------------------------------------------

typedef float v2f __attribute__((ext_vector_type(2)));
typedef float v8f __attribute__((ext_vector_type(8)));

#define NUM_TP   256     // total patterns
#define ML       6       // max pattern length
#define NDOCS    64
#define SEQ      128
#define DIM      300
#define PT       16      // patterns per block
#define ROWS     (PT*2*ML)   // 192 diag rows per block (consecutive)
#define KC       60      // D staged per LDS chunk (300 = 5*60, 60 = 15*4)
#define NWAVES   6
#define TBLOCK   (NWAVES*32) // 192 threads: 6 waves x 2 N-tiles = 12 N-tiles
#define NEGV     (-100.0f)
#define HID      100

__device__ __forceinline__ float sigmoidf(float x) {
    return 1.0f / (1.0f + __expf(-x));
}

__global__ __launch_bounds__(TBLOCK)
void sopa_fused(const float* __restrict__ emb,    // (64,128,300)
                const float* __restrict__ diags,  // (3072,300)
                const float* __restrict__ bias,   // (3072)
                const float* __restrict__ eps,    // (256,5)
                float* __restrict__ scores)       // (64,256)
{
    __shared__ float diag_lds[ROWS * KC];   // 45 KB
    __shared__ float emb_lds[16 * KC];      // 3.75 KB
    __shared__ float t_lds[16 * ROWS];      // 12 KB  sigmoid(T) tile

    const int ptile = blockIdx.x;           // 0..15 pattern tile
    const int doc   = blockIdx.y;           // 0..63
    const int tid   = threadIdx.x;
    const int wave  = tid >> 5;
    const int lane  = tid & 31;
    const int ln    = lane & 15;            // N index / M index within half-wave
    const int hi    = lane >> 4;            // K-half select for A/B frags
    const int rbase = ptile * ROWS;         // first global diag row of this tile

    // per-lane bias for the two N-tiles this wave owns (C/D layout: N = lane&15)
    const int nt0 = wave * 2, nt1 = wave * 2 + 1;
    const float bias0 = bias[rbase + nt0 * 16 + ln];
    const float bias1 = bias[rbase + nt1 * 16 + ln];

    // ---- scan state: thread p (<16) owns local pattern p, kept in registers
    const float sig05 = 1.0f / (1.0f + __expf(-0.5f));  // sigmoid(SCALE)
    float hreg[ML], epsv[ML - 1], sc = NEGV;
    hreg[0] = 0.0f;
    #pragma unroll
    for (int j = 1; j < ML; ++j) hreg[j] = NEGV;
    {
        const int pg = ptile * PT + (tid & (PT - 1));
        #pragma unroll
        for (int j = 0; j < ML - 1; ++j)
            epsv[j] = sig05 * sigmoidf(eps[pg * (ML - 1) + j]);
    }

    for (int c = 0; c < SEQ / 16; ++c) {            // 8 chunks of 16 tokens
        v8f acc0 = {}; v8f acc1 = {};               // two N-tiles per wave

        for (int dk = 0; dk < DIM; dk += KC) {      // 5 K chunks of 60
            __syncthreads();                         // LDS reuse guard
            // stage emb chunk (16 tokens x KC)
            for (int i = tid; i < 16 * KC; i += TBLOCK) {
                const int t = i / KC, k = i % KC;
                emb_lds[i] = emb[(doc * SEQ + c * 16 + t) * DIM + dk + k];
            }
            // stage diag chunk (192 rows x KC) + prefetch next chunk
            for (int i = tid; i < ROWS * KC; i += TBLOCK) {
                const int r = i / KC, k = i % KC;
                diag_lds[i] = diags[(rbase + r) * DIM + dk + k];
            }
            if (dk + KC < DIM)
                __builtin_prefetch(&diags[(rbase + (tid % ROWS)) * DIM + dk + KC], 0, 1);
            __syncthreads();

            // 15 K-steps of the native fp32 WMMA (16x16x4)
            #pragma unroll
            for (int ks = 0; ks < KC; ks += 4) {
                // A frag: lanes 0-15 -> K = ks..ks+1, lanes 16-31 -> K = ks+2..ks+3
                const v2f a  = *(const v2f*)&emb_lds[ln * KC + ks + 2 * hi];
                const v2f b0 = *(const v2f*)&diag_lds[(nt0 * 16 + ln) * KC + ks + 2 * hi];
                const v2f b1 = *(const v2f*)&diag_lds[(nt1 * 16 + ln) * KC + ks + 2 * hi];
                acc0 = __builtin_amdgcn_wmma_f32_16x16x4_f32(
                        false, a, false, b0, (short)0, acc0, false, false);
                acc1 = __builtin_amdgcn_wmma_f32_16x16x4_f32(
                        false, a, false, b1, (short)0, acc1, false, false);
            }
        }

        // bias + sigmoid, transpose into t_lds[token][row]
        // C/D layout: vgpr j -> token j (lanes 0-15) / token j+8 (lanes 16-31)
        #pragma unroll
        for (int j = 0; j < 8; ++j) {
            const int tok = j + 8 * hi;
            t_lds[tok * ROWS + nt0 * 16 + ln] = sigmoidf(acc0[j] + bias0);
            t_lds[tok * ROWS + nt1 * 16 + ln] = sigmoidf(acc1[j] + bias1);
        }
        __syncthreads();

        // max-plus scan over the 16 tokens of this chunk; thread p = pattern p
        if (tid < PT) {
            for (int t = 0; t < 16; ++t) {
                const float* Tt = &t_lds[t * ROWS + tid * (2 * ML)];
                float ae[ML];
                ae[0] = hreg[0];
                #pragma unroll
                for (int j = 1; j < ML; ++j)
                    ae[j] = fmaxf(hreg[j], hreg[j - 1] + epsv[j - 1]);
                float hn[ML];
                hn[0] = fmaxf(0.0f, sig05 * (ae[0] + Tt[0]));       // restart / self-loop
                #pragma unroll
                for (int j = 1; j < ML; ++j) {
                    const float mainv = ae[j - 1] + Tt[ML + (j - 1)]; // Tt[1][j-1]
                    const float selfv = sig05 * (ae[j] + Tt[j]);      // Tt[0][j]
                    hn[j] = fmaxf(mainv, selfv);
                }
                #pragma unroll
                for (int j = 0; j < ML; ++j) hreg[j] = hn[j];
                sc = fmaxf(sc, hreg[ML - 1]);
            }
        }
        __syncthreads();
    }

    if (tid < PT)
        scores[doc * NUM_TP + ptile * PT + tid] = sc;
}

// ---------------------------------------------------------------------------
// Stage 2: scores (64,256) -> relu(s@w1+b1) -> @w2+b2 -> log_softmax
// ---------------------------------------------------------------------------
__global__ __launch_bounds__(128)
void sopa_mlp(const float* __restrict__ scores, const float* __restrict__ w1,
              const float* __restrict__ b1, const float* __restrict__ w2,
              const float* __restrict__ b2, float* __restrict__ out)
{
    __shared__ float s_lds[NUM_TP];
    __shared__ float hid[HID];
    __shared__ float o[2];
    const int b = blockIdx.x, tid = threadIdx.x;

    for (int i = tid; i < NUM_TP; i += 128) s_lds[i] = scores[b * NUM_TP + i];
    __syncthreads();

    if (tid < HID) {
        float acc = b1[tid];
        for (int i = 0; i < NUM_TP; ++i) acc += s_lds[i] * w1[i * HID + tid];
        hid[tid] = fmaxf(acc, 0.0f);
    }
    __syncthreads();

    if (tid < 2) {
        float acc = b2[tid];
        for (int i = 0; i < HID; ++i) acc += hid[i] * w2[i * 2 + tid];
        o[tid] = acc;
    }
    __syncthreads();

    if (tid == 0) {
        const float m = fmaxf(o[0], o[1]);
        const float lse = m + __logf(__expf(o[0] - m) + __expf(o[1] - m));
        out[b * 2 + 0] = o[0] - lse;
        out[b * 2 + 1] = o[1] - lse;
    }
}

extern "C" void kernel_launch(void* const* d_in, const int* in_sizes, int n_in,
                              void* d_out, int out_size, void* d_ws, size_t ws_size,
                              hipStream_t stream)
{
    const float* emb   = (const float*)d_in[0];  // doc_embs (64,128,300)
    const float* diags = (const float*)d_in[1];  // (3072,300)
    const float* bias  = (const float*)d_in[2];  // (3072,1)
    const float* eps   = (const float*)d_in[3];  // (256,5)
    const float* w1    = (const float*)d_in[4];  // (256,100)
    const float* b1    = (const float*)d_in[5];  // (100)
    const float* w2    = (const float*)d_in[6];  // (100,2)
    const float* b2    = (const float*)d_in[7];  // (2)

    float* scores = (float*)d_ws;                // 64*256 floats = 64 KB

    dim3 grid(NUM_TP / PT, NDOCS);               // 16 x 64 = 1024 blocks
    sopa_fused<<<grid, TBLOCK, 0, stream>>>(emb, diags, bias, eps, scores);
    sopa_mlp<<<NDOCS, 128, 0, stream>>>(scores, w1, b1, w2, b2, (float*)d_out);
}